// Encoder_48739288875184
// MI455X (gfx1250) — compile-verified
//
#include <hip/hip_runtime.h>

typedef __attribute__((ext_vector_type(2))) float v2f;
typedef __attribute__((ext_vector_type(8))) float v8f;

#define INC 256
#define OUTC 64
#define SCALE_F 1.8f
#define EPS_F 1e-12f

// ---------------------------------------------------------------------------
// Degree / normalization helpers
// ---------------------------------------------------------------------------
__global__ void deg_init_kernel(unsigned* __restrict__ deg, int n) {
    int i = blockIdx.x * blockDim.x + threadIdx.x;
    if (i < n) deg[i] = 1u;  // self-loop
}

__global__ void deg_count_kernel(const int* __restrict__ dst,
                                 unsigned* __restrict__ deg, int e) {
    int i = blockIdx.x * blockDim.x + threadIdx.x;
    if (i < e) atomicAdd(&deg[dst[i]], 1u);
}

__global__ void dinv_kernel(const unsigned* __restrict__ deg,
                            float* __restrict__ dinv, int n) {
    int i = blockIdx.x * blockDim.x + threadIdx.x;
    if (i < n) dinv[i] = rsqrtf((float)deg[i]);
}

// ---------------------------------------------------------------------------
// Dense GEMM on the WMMA pipe: out[n,64] = x[n,256] @ w.T + bias
// fp32 A/B/C via V_WMMA_F32_16X16X4_F32; one wave computes a 16x64 tile.
// A 16x4 layout: lanes 0-15 hold M=0..15 K={0,1}; lanes 16-31 hold K={2,3}.
// B 4x16 layout mirrors A with N across lanes. C/D: VGPR r -> rows r / r+8.
// Out-of-range rows only pollute D rows that the store guard discards, so
// the tail needs only a pointer clamp (no value masking in the hot loop).
// ---------------------------------------------------------------------------
__global__ __launch_bounds__(256) void gemm_wmma_f32(
    const float* __restrict__ x, const float* __restrict__ w,
    const float* __restrict__ bias, float* __restrict__ out, int n)
{
    __shared__ float w_lds[OUTC * INC];  // 64 KB, row-major w[o][i]
    for (int i = threadIdx.x; i < OUTC * INC; i += 256)
        w_lds[i] = w[i];
    __syncthreads();

    const int lane    = threadIdx.x & 31;
    const int wave    = threadIdx.x >> 5;
    const int m       = lane & 15;           // A row-in-tile / B col-in-group
    const int khalf   = (lane >> 4) << 1;    // 0 or 2
    const int rowBase = blockIdx.x * 128 + wave * 16;
    const int arow    = rowBase + m;
    const float* xrow = x + (size_t)(arow < n ? arow : 0) * INC;

    float bvals[4];
    #pragma unroll
    for (int g = 0; g < 4; ++g) bvals[g] = bias[g * 16 + m];

    v8f acc[4] = {};
    for (int kk = 0; kk < INC; kk += 4) {
        const int k = kk + khalf;
        v2f a = *(const v2f*)(xrow + k);
        #pragma unroll
        for (int g = 0; g < 4; ++g) {
            const int col = g * 16 + m;
            v2f b = *(const v2f*)(w_lds + col * INC + k);
            acc[g] = __builtin_amdgcn_wmma_f32_16x16x4_f32(
                false, a, false, b, (short)0, acc[g], false, false);
        }
    }

    const int half = lane >> 4;
    #pragma unroll
    for (int r = 0; r < 8; ++r) {
        const int orow = rowBase + r + half * 8;
        if (orow < n) {
            #pragma unroll
            for (int g = 0; g < 4; ++g) {
                const int col = g * 16 + m;
                out[(size_t)orow * OUTC + col] = acc[g][r] + bvals[g];
            }
        }
    }
}

// ---------------------------------------------------------------------------
// Per-row L2 normalize * SCALE, one wave32 per row (2 channels per lane)
// ---------------------------------------------------------------------------
__global__ __launch_bounds__(256) void l2norm_scale_kernel(float* __restrict__ h, int n) {
    const int lane = threadIdx.x & 31;
    const int wave = threadIdx.x >> 5;
    const int row  = blockIdx.x * 8 + wave;
    if (row >= n) return;
    float2* p = (float2*)(h + (size_t)row * OUTC);
    float2 v = p[lane];
    float s = v.x * v.x + v.y * v.y;
    #pragma unroll
    for (int off = 16; off > 0; off >>= 1)
        s += __shfl_xor(s, off, 32);
    float nrm = fmaxf(sqrtf(s), EPS_F);
    float sc  = SCALE_F / nrm;
    v.x *= sc;
    v.y *= sc;
    p[lane] = v;
}

// ---------------------------------------------------------------------------
// Propagation: self-loop init, then edge scatter with f32 atomics (hit L2)
// ---------------------------------------------------------------------------
__global__ __launch_bounds__(256) void prop_init_kernel(
    const float* __restrict__ h1, const float* __restrict__ h2,
    const float* __restrict__ dinv,
    float* __restrict__ out_h1, float* __restrict__ out_h2, int n)
{
    int i = blockIdx.x * blockDim.x + threadIdx.x;
    if (i >= n * OUTC) return;
    int row = i >> 6;
    float w = dinv[row];
    w *= w;
    out_h1[i] = h1[i] * w;
    out_h2[i] = h2[i] * w;
}

__global__ __launch_bounds__(256) void prop_scatter_kernel(
    const int* __restrict__ src, const int* __restrict__ dst,
    const float* __restrict__ dinv,
    const float* __restrict__ h1, const float* __restrict__ h2,
    float* __restrict__ out_h1, float* __restrict__ out_h2, int e)
{
    long long t = (long long)blockIdx.x * blockDim.x + threadIdx.x;
    if (t >= (long long)e * 16) return;
    int eidx = (int)(t >> 4);
    int c    = (int)(t & 15) * 4;
    int s = src[eidx];
    int d = dst[eidx];
    float w = dinv[s] * dinv[d];
    float4 m1 = *(const float4*)(h1 + (size_t)s * OUTC + c);
    float4 m2 = *(const float4*)(h2 + (size_t)s * OUTC + c);
    float* o1 = out_h1 + (size_t)d * OUTC + c;
    float* o2 = out_h2 + (size_t)d * OUTC + c;
    atomicAdd(o1 + 0, m1.x * w);
    atomicAdd(o1 + 1, m1.y * w);
    atomicAdd(o1 + 2, m1.z * w);
    atomicAdd(o1 + 3, m1.w * w);
    atomicAdd(o2 + 0, m2.x * w);
    atomicAdd(o2 + 1, m2.y * w);
    atomicAdd(o2 + 2, m2.z * w);
    atomicAdd(o2 + 3, m2.w * w);
}

// ---------------------------------------------------------------------------
extern "C" void kernel_launch(void* const* d_in, const int* in_sizes, int n_in,
                              void* d_out, int out_size, void* d_ws, size_t ws_size,
                              hipStream_t stream)
{
    const float* x  = (const float*)d_in[0];
    const int*   ei = (const int*)d_in[1];
    const float* w1 = (const float*)d_in[2];
    const float* b1 = (const float*)d_in[3];
    const float* w2 = (const float*)d_in[4];
    const float* b2 = (const float*)d_in[5];

    const int n = in_sizes[0] / INC;   // 100000
    const int e = in_sizes[1] / 2;     // 1600000
    const int* src = ei;
    const int* dst = ei + e;

    // Workspace carve-up (≈52.0 MB)
    char* ws = (char*)d_ws;
    unsigned* deg = (unsigned*)ws;  ws += (((size_t)n * 4) + 255) & ~(size_t)255;
    float* dinv   = (float*)ws;     ws += (((size_t)n * 4) + 255) & ~(size_t)255;
    float* h1     = (float*)ws;     ws += (size_t)n * OUTC * 4;
    float* h2     = (float*)ws;

    // Output: reference returns (h_branch2, x_branch1) concatenated
    float* out_h2 = (float*)d_out;
    float* out_h1 = out_h2 + (size_t)n * OUTC;

    deg_init_kernel<<<(n + 255) / 256, 256, 0, stream>>>(deg, n);
    deg_count_kernel<<<(e + 255) / 256, 256, 0, stream>>>(dst, deg, e);
    dinv_kernel<<<(n + 255) / 256, 256, 0, stream>>>(deg, dinv, n);

    gemm_wmma_f32<<<(n + 127) / 128, 256, 0, stream>>>(x, w1, b1, h1, n);
    gemm_wmma_f32<<<(n + 127) / 128, 256, 0, stream>>>(x, w2, b2, h2, n);

    l2norm_scale_kernel<<<(n + 7) / 8, 256, 0, stream>>>(h2, n);

    prop_init_kernel<<<((size_t)n * OUTC + 255) / 256, 256, 0, stream>>>(
        h1, h2, dinv, out_h1, out_h2, n);

    long long tot = (long long)e * 16;
    prop_scatter_kernel<<<(unsigned)((tot + 255) / 256), 256, 0, stream>>>(
        src, dst, dinv, h1, h2, out_h1, out_h2, e);
}